// DifferentiablePhysicsEngine_62113817034787
// MI455X (gfx1250) — compile-verified
//
#include <hip/hip_runtime.h>
#include <hip/hip_bf16.h>
#include <math.h>

typedef __attribute__((ext_vector_type(2))) float v2f;
typedef __attribute__((ext_vector_type(8))) float v8f;

#define EPSF      1e-6f
#define NB        8
#define NS        512
#define NAT       2048                 // atoms per batch (S*A)
#define LJ_BLOCKS 256                  // 2048 waves; wave = half-strip (64 tiles)
#define HB_BLOCKS 32                   // 256 waves; wave = one strip (32 tiles)
#define C_EXP     (-18.033688f)        // -12.5 * log2(e)

// ---------------------------------------------------------------------------
// Kernel 1: per-atom padded coords (x,y,z,0) + (squared norm, fmask)
// ---------------------------------------------------------------------------
__global__ void dpe_prep(const float* __restrict__ coords,
                         const float* __restrict__ mask,
                         float4* __restrict__ x4,
                         float2* __restrict__ pair2) {
    int idx = blockIdx.x * blockDim.x + threadIdx.x;   // 0 .. 16383
    if (idx >= NB * NAT) return;
    int b = idx >> 11;
    int n = idx & (NAT - 1);
    const float* p = coords + (size_t)idx * 3;
    float x = p[0], y = p[1], z = p[2];
    float fm = mask[b * NS + (n >> 2)];
    x4[idx]    = make_float4(x, y, z, 0.0f);
    pair2[idx] = make_float2(x * x + y * y + z * z, fm);
}

// ---------------------------------------------------------------------------
// Kernel 2: small scalar terms (mask sums, motor, rg) — one block, 8 waves.
// ---------------------------------------------------------------------------
__global__ void dpe_scalars(const float* __restrict__ coords,
                            const float* __restrict__ motors,
                            const float* __restrict__ mask,
                            float* __restrict__ sc) {
    __shared__ float s_msum[8], s_msq[8], s_mn[8], s_md[8], s_rg[8];
    int wave = threadIdx.x >> 5;
    int lane = threadIdx.x & 31;
    int b = wave;
    const float* mb = mask + b * NS;

    float msum = 0.f, msq = 0.f, sx = 0.f, sy = 0.f, sz = 0.f;
    for (int s = lane; s < NS; s += 32) {
        float m = mb[s];
        msum += m; msq += m * m;
        const float* ca = coords + (size_t)(b * NAT + s * 4 + 1) * 3;
        sx += ca[0] * m; sy += ca[1] * m; sz += ca[2] * m;
    }
    for (int off = 16; off > 0; off >>= 1) {
        msum += __shfl_xor(msum, off, 32);
        msq  += __shfl_xor(msq,  off, 32);
        sx   += __shfl_xor(sx,   off, 32);
        sy   += __shfl_xor(sy,   off, 32);
        sz   += __shfl_xor(sz,   off, 32);
    }
    float inv = 1.0f / (msum + EPSF);
    float mx = sx * inv, my = sy * inv, mz = sz * inv;

    float dsum = 0.f;
    for (int s = lane; s < NS; s += 32) {
        float m = mb[s];
        const float* ca = coords + (size_t)(b * NAT + s * 4 + 1) * 3;
        float dx = ca[0] - mx, dy = ca[1] - my, dz = ca[2] - mz;
        dsum += (dx * dx + dy * dy + dz * dz) * m;
    }
    float mn = 0.f, md = 0.f;
    const float* mo = motors + (size_t)b * NS * 8;
    for (int s = lane; s < NS - 1; s += 32) {
        float d = 0.f;
        #pragma unroll
        for (int c = 0; c < 8; ++c) {
            float t = mo[(s + 1) * 8 + c] - mo[s * 8 + c];
            d += t * t;
        }
        float mm = mb[s + 1] * mb[s];
        mn += d * mm; md += mm;
    }
    for (int off = 16; off > 0; off >>= 1) {
        dsum += __shfl_xor(dsum, off, 32);
        mn   += __shfl_xor(mn,   off, 32);
        md   += __shfl_xor(md,   off, 32);
    }
    if (lane == 0) {
        s_msum[wave] = msum; s_msq[wave] = msq;
        s_mn[wave] = mn; s_md[wave] = md;
        s_rg[wave] = dsum * inv;
    }
    __syncthreads();
    if (threadIdx.x == 0) {
        float pd = 0.f, ms = 0.f, motn = 0.f, motd = 0.f, rg = 0.f;
        for (int i = 0; i < 8; ++i) {
            pd   += 16.0f * s_msum[i] * s_msum[i] - 4.0f * s_msq[i];
            ms   += s_msum[i];
            motn += s_mn[i]; motd += s_md[i];
            rg   += s_rg[i];
        }
        sc[0] = pd; sc[1] = ms; sc[2] = motn; sc[3] = motd; sc[4] = rg * 0.125f;
    }
}

// ---------------------------------------------------------------------------
// Kernel 3: LJ clash — strip-parallel Gram via V_WMMA_F32_16X16X4_F32.
// Wave holds A operand + i-side norms in registers; sweeps 64 j-tiles.
// ---------------------------------------------------------------------------
__global__ void __launch_bounds__(256)
dpe_lj(const float2* __restrict__ X2,        // (x,y)|(z,0) pairs of padded coords
       const float2* __restrict__ pair2,
       float* __restrict__ part) {
    int wave  = threadIdx.x >> 5;
    int lane  = threadIdx.x & 31;
    int gw    = blockIdx.x * 8 + wave;       // 0..2047
    int strip = gw >> 1;                     // 0..1023 : (b, ti)
    int halfj = gw & 1;
    int b     = strip >> 7;
    int i0    = (strip & 127) << 4;
    int row   = lane & 15;
    int half  = lane >> 4;                   // 0: K=0,1   1: K=2,pad
    int moff  = half << 3;
    size_t base = (size_t)b * NAT;

    float2 av = X2[(base + i0 + row) * 2 + half];
    v2f Av; Av[0] = av.x; Av[1] = av.y;

    float nrm_i[8], fm_i[8];
    #pragma unroll
    for (int r = 0; r < 8; ++r) {
        float2 p = pair2[base + i0 + moff + r];
        nrm_i[r] = p.x; fm_i[r] = p.y;
    }

    float acc = 0.f;
    int tj0 = halfj << 6;
    for (int tj = tj0; tj < tj0 + 64; ++tj) {
        int j0 = tj << 4;
        float2 bv = X2[(base + j0 + row) * 2 + half];
        v2f Bv; Bv[0] = bv.x; Bv[1] = bv.y;
        v8f C = {};
        C = __builtin_amdgcn_wmma_f32_16x16x4_f32(
                false, Av, false, Bv, (short)0, C, false, false);
        int j = j0 + (lane & 15);
        float2 pj = pair2[base + j];
        #pragma unroll
        for (int r = 0; r < 8; ++r) {
            int i = i0 + moff + r;
            float d2 = fmaxf(nrm_i[r] + pj.x - 2.0f * C[r], 0.0f) + EPSF;
            float rr = __builtin_amdgcn_sqrtf(d2);     // v_sqrt_f32
            float ov = fmaxf(3.0f - rr, 0.0f);
            float ov2 = ov * ov;
            float w = (i == j) ? 0.0f : fm_i[r] * pj.y;
            acc += (ov2 * ov2) * w;
        }
    }
    for (int off = 16; off > 0; off >>= 1) acc += __shfl_xor(acc, off, 32);
    __shared__ float s[8];
    if (lane == 0) s[wave] = acc;
    __syncthreads();
    if (threadIdx.x == 0) {
        float tsum = 0.f;
        for (int i = 0; i < 8; ++i) tsum += s[i];
        part[blockIdx.x] = tsum;
    }
}

// ---------------------------------------------------------------------------
// Kernel 4: HB term — strip-parallel Nc·Oc^T Gram via WMMA + fast exp2.
// ---------------------------------------------------------------------------
__global__ void __launch_bounds__(256)
dpe_hb(const float2* __restrict__ X2,
       const float2* __restrict__ pair2,
       const float* __restrict__ mask,
       float* __restrict__ part) {
    int wave = threadIdx.x >> 5;
    int lane = threadIdx.x & 31;
    int gw   = blockIdx.x * 8 + wave;        // 0..255 : (b, ti)
    int b    = gw >> 5;
    int i0   = (gw & 31) << 4;
    int row  = lane & 15;
    int half = lane >> 4;
    int moff = half << 3;
    size_t base = (size_t)b * NAT;

    float2 av = X2[(base + (size_t)(i0 + row) * 4) * 2 + half];   // Nc: atom 0
    v2f Av; Av[0] = av.x; Av[1] = av.y;

    float nrmN[8], mi[8];
    #pragma unroll
    for (int r = 0; r < 8; ++r) {
        nrmN[r] = pair2[base + (size_t)(i0 + moff + r) * 4].x;
        mi[r]   = mask[b * NS + i0 + moff + r];
    }

    float acc = 0.f;
    for (int tj = 0; tj < 32; ++tj) {
        int j0 = tj << 4;
        float2 bv = X2[(base + (size_t)(j0 + row) * 4 + 3) * 2 + half]; // Oc: atom 3
        v2f Bv; Bv[0] = bv.x; Bv[1] = bv.y;
        v8f C = {};
        C = __builtin_amdgcn_wmma_f32_16x16x4_f32(
                false, Av, false, Bv, (short)0, C, false, false);
        int j = j0 + (lane & 15);
        float nrmO = pair2[base + (size_t)j * 4 + 3].x;
        float mj   = mask[b * NS + j];
        #pragma unroll
        for (int r = 0; r < 8; ++r) {
            int i = i0 + moff + r;
            float d2   = fmaxf(nrmN[r] + nrmO - 2.0f * C[r], 0.0f);
            float dist = __builtin_amdgcn_sqrtf(d2);   // v_sqrt_f32
            float dd   = dist - 2.9f;
            float h    = __builtin_amdgcn_exp2f(C_EXP * dd * dd); // v_exp_f32
            int  ad    = i - j; ad = ad < 0 ? -ad : ad;
            float w    = (ad > 2) ? mi[r] * mj : 0.0f;
            acc += h * w;
        }
    }
    for (int off = 16; off > 0; off >>= 1) acc += __shfl_xor(acc, off, 32);
    __shared__ float s[8];
    if (lane == 0) s[wave] = acc;
    __syncthreads();
    if (threadIdx.x == 0) {
        float tsum = 0.f;
        for (int i = 0; i < 8; ++i) tsum += s[i];
        part[blockIdx.x] = tsum;
    }
}

// ---------------------------------------------------------------------------
// Kernel 5: deterministic final reduction + combine
// ---------------------------------------------------------------------------
__global__ void dpe_final(const float* __restrict__ sc,
                          const float* __restrict__ lj_part,
                          const float* __restrict__ hb_part,
                          float* __restrict__ out) {
    __shared__ float sh[256];
    float a = 0.f, h = 0.f;
    for (int i = threadIdx.x; i < LJ_BLOCKS; i += 256) a += lj_part[i];
    for (int i = threadIdx.x; i < HB_BLOCKS; i += 256) h += hb_part[i];

    sh[threadIdx.x] = a; __syncthreads();
    for (int s = 128; s > 0; s >>= 1) {
        if (threadIdx.x < s) sh[threadIdx.x] += sh[threadIdx.x + s];
        __syncthreads();
    }
    float lj_sum = sh[0]; __syncthreads();

    sh[threadIdx.x] = h; __syncthreads();
    for (int s = 128; s > 0; s >>= 1) {
        if (threadIdx.x < s) sh[threadIdx.x] += sh[threadIdx.x + s];
        __syncthreads();
    }
    float hb_sum = sh[0];

    if (threadIdx.x == 0) {
        float lj    = lj_sum / (sc[0] + EPSF);
        float hb    = -hb_sum / (sc[1] + EPSF);
        float motor = sc[2] / (sc[3] + EPSF);
        float rg    = sc[4];
        out[0] = lj + 0.5f * hb + 0.1f * motor + 0.05f * rg;
    }
}

// ---------------------------------------------------------------------------
extern "C" void kernel_launch(void* const* d_in, const int* in_sizes, int n_in,
                              void* d_out, int out_size, void* d_ws, size_t ws_size,
                              hipStream_t stream) {
    const float* coords = (const float*)d_in[0];   // (8,512,4,3) f32
    const float* motors = (const float*)d_in[1];   // (8,512,8)  f32
    const float* mask   = (const float*)d_in[2];   // (8,512)    f32

    float*  ws      = (float*)d_ws;
    float4* x4      = (float4*)ws;                 // 16384 float4 = 65536 f
    float2* pair2   = (float2*)(ws + 4 * NB * NAT);// 16384 float2 = 32768 f
    float*  sc      = ws + 6 * NB * NAT;           // 8 scalars
    float*  lj_part = sc + 8;                      // LJ_BLOCKS floats
    float*  hb_part = lj_part + LJ_BLOCKS;         // HB_BLOCKS floats
    float*  out     = (float*)d_out;

    dpe_prep   <<<64, 256, 0, stream>>>(coords, mask, x4, pair2);
    dpe_scalars<<<1, 256, 0, stream>>>(coords, motors, mask, sc);
    dpe_lj     <<<LJ_BLOCKS, 256, 0, stream>>>((const float2*)x4, pair2, lj_part);
    dpe_hb     <<<HB_BLOCKS, 256, 0, stream>>>((const float2*)x4, pair2, mask, hb_part);
    dpe_final  <<<1, 256, 0, stream>>>(sc, lj_part, hb_part, out);
}